// HebbianMemory_87316685128046
// MI455X (gfx1250) — compile-verified
//
#include <hip/hip_runtime.h>

// ---------------------------------------------------------------------------
// Hebbian memory (decaying outer-product state) restructured into chunk-
// parallel WMMA GEMMs for MI455X (gfx1250, wave32, V_WMMA_F32_16X16X4_F32).
//
//   A_t = lambda*A_{t-1} + eta*k_t v_t^T ;  y_t = q_t A_t
// Chunked (C=256, 0-based tau in chunk):
//   y_tau = lambda^{tau+1} q_tau A_start
//         + eta * sum_{s<=tau} lambda^{tau-s} (q_tau . k_s) v_s
//   dA_chunk = eta * sum_s lambda^{C-1-s} k_s v_s^T
//   A_next_start = lambda^C A_start + dA_chunk         (tiny 16-step scan)
// ---------------------------------------------------------------------------

#define DIM     256
#define BATCH   8
#define SEQ     4096
#define CHUNK   256
#define NCHUNK  (SEQ / CHUNK)          // 16
#define ETA     0.1f
#define LAMBDA  0.9f
#define LAMBDA_POW_CHUNK 1.9323349e-12f  // 0.9^256

typedef __attribute__((ext_vector_type(2))) float v2f;
typedef __attribute__((ext_vector_type(8))) float v8f;

// D = A(16x4, f32) * B(4x16, f32) + C(16x16, f32)
// A layout: lane r(0..15)=row M, lane-group g + vgpr j -> K = 2g+j
// B layout: lane r=col N, lane-group g + vgpr j -> K = 2g+j
// C/D layout: vgpr i, lane-group g -> M = i+8g, lane r -> N
__device__ __forceinline__ v8f wmma_f32(v2f a, v2f b, v8f c) {
  return __builtin_amdgcn_wmma_f32_16x16x4_f32(
      /*neg_a=*/false, a, /*neg_b=*/false, b,
      /*c_mod=*/(short)0, c, /*reuse_a=*/false, /*reuse_b=*/false);
}

// ---------------------------------------------------------------------------
// out[m,n] = sum_d in[m,d]*W[n,d] + bias[n]   (torch Linear: x @ W^T + b)
// WG = 4 waves; wave w -> cols [64w, 64w+64); WG -> rows [16*blockIdx.x, +16)
// ---------------------------------------------------------------------------
__global__ __launch_bounds__(128) void proj_kernel(
    const float* __restrict__ in, const float* __restrict__ W,
    const float* __restrict__ bias, float* __restrict__ out) {
  const int m0   = blockIdx.x * 16;
  const int wave = threadIdx.x >> 5;
  const int lane = threadIdx.x & 31;
  const int g = lane >> 4, r = lane & 15;
  const int n0 = wave * 64;

  v8f acc[4] = {};
  const float* arow = in + (size_t)(m0 + r) * DIM + 2 * g;
#pragma unroll 4
  for (int kk = 0; kk < DIM; kk += 4) {
    v2f a = *(const v2f*)(arow + kk);      // x[m0+r, kk+2g .. +1] contiguous
#pragma unroll
    for (int nt = 0; nt < 4; ++nt) {
      // B[k][n] = W[n][k] -> row (n0+nt*16+r) of W, contiguous pair
      v2f b = *(const v2f*)(W + (size_t)(n0 + nt * 16 + r) * DIM + kk + 2 * g);
      acc[nt] = wmma_f32(a, b, acc[nt]);
    }
  }
#pragma unroll
  for (int nt = 0; nt < 4; ++nt) {
    const int n = n0 + nt * 16 + r;
    const float bv = bias[n];
#pragma unroll
    for (int i = 0; i < 8; ++i)
      out[(size_t)(m0 + i + 8 * g) * DIM + n] = acc[nt][i] + bv;
  }
}

// ---------------------------------------------------------------------------
// dA[b,c][m,n] = sum_s eta*lambda^{C-1-s} * K[b,cC+s,m] * V[b,cC+s,n]
// (K^T V with per-s decay weights). Grid: B*NCHUNK*16 WGs of 4 waves.
// ---------------------------------------------------------------------------
__global__ __launch_bounds__(128) void delta_kernel(
    const float* __restrict__ Kbuf, const float* __restrict__ Vbuf,
    float* __restrict__ dA) {
  __shared__ float lamPow[CHUNK];
  lamPow[threadIdx.x]       = __powf(LAMBDA, (float)threadIdx.x);
  lamPow[threadIdx.x + 128] = __powf(LAMBDA, (float)(threadIdx.x + 128));
  __syncthreads();

  const int idx = blockIdx.x;
  const int mb = idx & 15;
  const int c  = (idx >> 4) % NCHUNK;
  const int b  = idx / (16 * NCHUNK);
  const int m0 = mb * 16;
  const int wave = threadIdx.x >> 5;
  const int lane = threadIdx.x & 31;
  const int g = lane >> 4, r = lane & 15;
  const int n0 = wave * 64;

  const float* Kc = Kbuf + ((size_t)b * SEQ + (size_t)c * CHUNK) * DIM;
  const float* Vc = Vbuf + ((size_t)b * SEQ + (size_t)c * CHUNK) * DIM;

  v8f acc[4] = {};
#pragma unroll 2
  for (int s = 0; s < CHUNK; s += 4) {
    const int sA = s + 2 * g;              // reduction index for this lane-group
    v2f a;                                 // A[m][s] = coeff(s)*K[s][m]
    a.x = ETA * lamPow[CHUNK - 1 - sA] * Kc[(size_t)sA * DIM + m0 + r];
    a.y = ETA * lamPow[CHUNK - 2 - sA] * Kc[(size_t)(sA + 1) * DIM + m0 + r];
#pragma unroll
    for (int nt = 0; nt < 4; ++nt) {
      const int n = n0 + nt * 16 + r;
      v2f bf;                              // B[s][n] = V[s][n]
      bf.x = Vc[(size_t)sA * DIM + n];
      bf.y = Vc[(size_t)(sA + 1) * DIM + n];
      acc[nt] = wmma_f32(a, bf, acc[nt]);
    }
  }
  float* o = dA + (size_t)(b * NCHUNK + c) * DIM * DIM;
#pragma unroll
  for (int nt = 0; nt < 4; ++nt)
#pragma unroll
    for (int i = 0; i < 8; ++i)
      o[(size_t)(m0 + i + 8 * g) * DIM + n0 + nt * 16 + r] = acc[nt][i];
}

// ---------------------------------------------------------------------------
// Serial chunk scan (16 steps) over the D x D state per batch.
// Records each chunk's starting state; writes final state to d_out tail.
// ---------------------------------------------------------------------------
__global__ __launch_bounds__(256) void scan_kernel(
    const float* __restrict__ state_in, const float* __restrict__ dA,
    float* __restrict__ Astart, float* __restrict__ finalState) {
  const int idx = blockIdx.x * 256 + threadIdx.x;     // over B*D*D
  const int b   = idx / (DIM * DIM);
  const int rem = idx % (DIM * DIM);
  float A = state_in[idx];
#pragma unroll
  for (int c = 0; c < NCHUNK; ++c) {
    const size_t o = (size_t)(b * NCHUNK + c) * DIM * DIM + rem;
    Astart[o] = A;
    A = LAMBDA_POW_CHUNK * A + dA[o];
  }
  finalState[idx] = A;
}

// ---------------------------------------------------------------------------
// Y[b, cC+t, e] = lambda^{t+1} * Q[t,:] @ Astart
//              + eta * sum_{s<=t} lambda^{t-s} (Q[t,:].K[s,:]) V[s,e]
// WG = 4 waves handles rows [t0,t0+16) x all 256 cols of one (b,c).
// Each wave produces one masked 16x16 score block per 64-wide outer step,
// stages it via LDS (C-layout -> A-layout), then all waves consume it vs V.
// ---------------------------------------------------------------------------
__global__ __launch_bounds__(128) void y_kernel(
    const float* __restrict__ Qbuf, const float* __restrict__ Kbuf,
    const float* __restrict__ Vbuf, const float* __restrict__ Astart,
    float* __restrict__ Ybuf) {
  __shared__ float lamPow[CHUNK];
  __shared__ float Stile[4][256];
  lamPow[threadIdx.x]       = __powf(LAMBDA, (float)threadIdx.x);
  lamPow[threadIdx.x + 128] = __powf(LAMBDA, (float)(threadIdx.x + 128));
  __syncthreads();

  const int idx = blockIdx.x;
  const int tb = idx & 15;
  const int c  = (idx >> 4) % NCHUNK;
  const int b  = idx / (16 * NCHUNK);
  const int t0 = tb * 16;
  const int wave = threadIdx.x >> 5;
  const int lane = threadIdx.x & 31;
  const int g = lane >> 4, r = lane & 15;
  const int n0 = wave * 64;

  const size_t cb = ((size_t)b * SEQ + (size_t)c * CHUNK) * DIM;
  const float* Qc = Qbuf + cb;
  const float* Kc = Kbuf + cb;
  const float* Vc = Vbuf + cb;
  const float* As = Astart + (size_t)(b * NCHUNK + c) * DIM * DIM;

  v8f acc[4] = {};
  const float* qrow = Qc + (size_t)(t0 + r) * DIM + 2 * g;

  // --- inter-chunk term: lambda^{t+1} * Q @ Astart  (K-reduction over D) ---
  const float rowscale = lamPow[t0 + r] * LAMBDA;     // lambda^{(t0+r)+1}
#pragma unroll 4
  for (int kk = 0; kk < DIM; kk += 4) {
    v2f a = *(const v2f*)(qrow + kk);
    a.x *= rowscale; a.y *= rowscale;
#pragma unroll
    for (int nt = 0; nt < 4; ++nt) {
      const int n = n0 + nt * 16 + r;
      v2f bf;                                         // B[d][e] = Astart[d][e]
      bf.x = As[(size_t)(kk + 2 * g) * DIM + n];
      bf.y = As[(size_t)(kk + 2 * g + 1) * DIM + n];
      acc[nt] = wmma_f32(a, bf, acc[nt]);
    }
  }

  // --- intra-chunk causal term ---
  for (int sOuter = 0; sOuter <= t0; sOuter += 64) {
    const int s0 = sOuter + wave * 16;                // this wave's score block
    if (s0 <= t0 + 15) {                              // wave-uniform
      v8f p = {};
      const float* krow = Kc + (size_t)(s0 + r) * DIM + 2 * g;
#pragma unroll 4
      for (int kk = 0; kk < DIM; kk += 4) {           // P = Q K^T over d
        v2f a  = *(const v2f*)(qrow + kk);
        v2f bf = *(const v2f*)(krow + kk);            // B[d][s] = K[s][d]
        p = wmma_f32(a, bf, p);
      }
#pragma unroll
      for (int i = 0; i < 8; ++i) {                   // causal decay mask
        const int t = t0 + i + 8 * g;
        const int s = s0 + r;
        Stile[wave][(i + 8 * g) * 16 + r] =
            (s <= t) ? ETA * lamPow[t - s] * p[i] : 0.0f;
      }
    }
    __syncthreads();
#pragma unroll
    for (int sb = 0; sb < 4; ++sb) {
      const int sblk = sOuter + sb * 16;
      if (sblk > t0 + 15) continue;                   // WG-uniform skip
#pragma unroll
      for (int kkl = 0; kkl < 16; kkl += 4) {         // Y += Stile * V
        v2f a;                                        // A-layout read of Stile
        a.x = Stile[sb][r * 16 + kkl + 2 * g];
        a.y = Stile[sb][r * 16 + kkl + 2 * g + 1];
#pragma unroll
        for (int nt = 0; nt < 4; ++nt) {
          const int n = n0 + nt * 16 + r;
          v2f bf;
          bf.x = Vc[(size_t)(sblk + kkl + 2 * g) * DIM + n];
          bf.y = Vc[(size_t)(sblk + kkl + 2 * g + 1) * DIM + n];
          acc[nt] = wmma_f32(a, bf, acc[nt]);
        }
      }
    }
    __syncthreads();
  }

#pragma unroll
  for (int nt = 0; nt < 4; ++nt)
#pragma unroll
    for (int i = 0; i < 8; ++i)
      Ybuf[cb + (size_t)(t0 + i + 8 * g) * DIM + n0 + nt * 16 + r] = acc[nt][i];
}

// ---------------------------------------------------------------------------
extern "C" void kernel_launch(void* const* d_in, const int* in_sizes, int n_in,
                              void* d_out, int out_size, void* d_ws, size_t ws_size,
                              hipStream_t stream) {
  (void)in_sizes; (void)n_in; (void)out_size; (void)ws_size;
  const float* x  = (const float*)d_in[0];
  const float* st = (const float*)d_in[1];
  const float* Wk = (const float*)d_in[2];
  const float* bk = (const float*)d_in[3];
  const float* Wv = (const float*)d_in[4];
  const float* bv = (const float*)d_in[5];
  const float* Wq = (const float*)d_in[6];
  const float* bq = (const float*)d_in[7];
  const float* Wo = (const float*)d_in[8];
  const float* bo = (const float*)d_in[9];
  float* out = (float*)d_out;

  const size_t BLD = (size_t)BATCH * SEQ * DIM;              // 8,388,608
  const size_t AST = (size_t)BATCH * NCHUNK * DIM * DIM;     // 8,388,608
  float* ws     = (float*)d_ws;
  float* Kbuf   = ws;
  float* Vbuf   = Kbuf + BLD;
  float* Qbuf   = Vbuf + BLD;
  float* Ybuf   = Qbuf + BLD;
  float* dA     = Ybuf + BLD;
  float* AstartB = dA + AST;

  const dim3 blk(128);
  const int projGrid = (BATCH * SEQ) / 16;                   // 2048 WGs
  proj_kernel<<<projGrid, blk, 0, stream>>>(x, Wk, bk, Kbuf);
  proj_kernel<<<projGrid, blk, 0, stream>>>(x, Wv, bv, Vbuf);
  proj_kernel<<<projGrid, blk, 0, stream>>>(x, Wq, bq, Qbuf);

  const int dGrid = BATCH * NCHUNK * (DIM / 16);             // 2048 WGs
  delta_kernel<<<dGrid, blk, 0, stream>>>(Kbuf, Vbuf, dA);

  scan_kernel<<<(BATCH * DIM * DIM) / 256, 256, 0, stream>>>(st, dA, AstartB,
                                                             out + BLD);

  const int yGrid = BATCH * NCHUNK * (CHUNK / 16);           // 2048 WGs
  y_kernel<<<yGrid, blk, 0, stream>>>(Qbuf, Kbuf, Vbuf, AstartB, Ybuf);

  proj_kernel<<<projGrid, blk, 0, stream>>>(Ybuf, Wo, bo, out);
}